// GCN_25546465477207
// MI455X (gfx1250) — compile-verified
//
#include <hip/hip_runtime.h>
#include <hip/hip_bf16.h>
#include <math.h>

// ---------------------------------------------------------------------------
// GCN forward on MI455X (gfx1250, wave32).
//   N=50000 nodes, E=800000 edges, 256 -> 64 -> 40 channels.
// GEMMs use V_WMMA_F32_16X16X4_F32 (full fp32 precision, matrix pipe).
// Edge aggregation is the bandwidth-bound part (~0.7GB traffic): coalesced
// per-edge channel vectors + global_atomic_add_f32 scatter.
// ---------------------------------------------------------------------------

typedef __attribute__((ext_vector_type(2))) float v2f;
typedef __attribute__((ext_vector_type(8))) float v8f;

#define IN_C 256
#define HID_C 64
#define OUT_C 40
#define OUT_P 48   // OUT_C padded to multiple of 16 for WMMA tiles

// ---------------- degree / normalization ----------------

__global__ __launch_bounds__(256) void k_init_deg(float* __restrict__ deg, int n) {
    int i = blockIdx.x * blockDim.x + threadIdx.x;
    if (i < n) deg[i] = 1.0f;   // self-loop contribution
}

__global__ __launch_bounds__(256) void k_deg_edges(const int* __restrict__ dst,
                                                   float* __restrict__ deg, int e) {
    int i = blockIdx.x * blockDim.x + threadIdx.x;
    if (i < e) atomicAdd(&deg[dst[i]], 1.0f);
}

__global__ __launch_bounds__(256) void k_deg_to_dinv(float* __restrict__ deg, int n) {
    int i = blockIdx.x * blockDim.x + threadIdx.x;
    if (i < n) deg[i] = rsqrtf(deg[i]);   // deg >= 1 always (self-loops)
}

// ---------------- GEMM 1: hlin[M,64] = x[M,256] @ W1[256,64] ----------------
// One wave per 16x16 output tile; K-loop of 64 x V_WMMA_F32_16X16X4_F32.
// A frag (16x4 f32): lane l holds rows M=l%16, K = 2*(l/16)+{0,1}  (v2f).
// B frag (4x16 f32): lane l holds col N=l%16, K = 2*(l/16)+{0,1}  (v2f).
// C/D (16x16 f32):   VGPR v -> M = v + 8*(l/16), N = l%16.

__global__ __launch_bounds__(256) void k_gemm1_wmma(const float* __restrict__ x,
                                                    const float* __restrict__ W1,
                                                    float* __restrict__ hlin, int M) {
    const int NT = HID_C / 16;                         // 4 col tiles
    int wave = blockIdx.x * (blockDim.x >> 5) + (threadIdx.x >> 5);
    int lane = threadIdx.x & 31;
    int mt = wave / NT, nt = wave % NT;
    if (mt * 16 >= M) return;                          // wave-uniform guard

    int row  = mt * 16 + (lane & 15);
    int kg   = (lane >> 4) * 2;                        // 0 or 2
    int ncol = nt * 16 + (lane & 15);

    v8f acc = {};
    #pragma unroll 4
    for (int k0 = 0; k0 < IN_C; k0 += 4) {
        v2f a = *(const v2f*)(x + (size_t)row * IN_C + k0 + kg);
        v2f b;
        b.x = W1[(size_t)(k0 + kg + 0) * HID_C + ncol];
        b.y = W1[(size_t)(k0 + kg + 1) * HID_C + ncol];
        acc = __builtin_amdgcn_wmma_f32_16x16x4_f32(
                  false, a, false, b, (short)0, acc, false, false);
    }
    int half = (lane >> 4) * 8;
    #pragma unroll
    for (int v = 0; v < 8; ++v)
        hlin[(size_t)(mt * 16 + half + v) * HID_C + nt * 16 + (lane & 15)] = acc[v];
}

// ---------------- layer-1 aggregation ----------------

// self-loop init: out1[i,c] = hlin[i,c] * dinv[i]^2
__global__ __launch_bounds__(256) void k_selfloop1(const float* __restrict__ hlin,
                                                   const float* __restrict__ dinv,
                                                   float* __restrict__ out1, int n) {
    int idx = blockIdx.x * blockDim.x + threadIdx.x;
    int i = idx >> 6;
    if (i < n) {
        float d = dinv[i];
        out1[idx] = hlin[idx] * d * d;
    }
}

// edge scatter: out1[dst,c] += hlin[src,c] * dinv[src]*dinv[dst]
// 64 consecutive lanes cover one edge's channels -> coalesced 256B gather.
__global__ __launch_bounds__(256) void k_agg1(const float* __restrict__ hlin,
                                              const int* __restrict__ src,
                                              const int* __restrict__ dst,
                                              const float* __restrict__ dinv,
                                              float* __restrict__ out1, int e) {
    unsigned idx = blockIdx.x * blockDim.x + threadIdx.x;
    int ed = idx >> 6, c = idx & 63;
    if (ed >= e) return;
    int s = src[ed], d = dst[ed];
    float nrm = dinv[s] * dinv[d];
    atomicAdd(out1 + (size_t)d * HID_C + c, hlin[(size_t)s * HID_C + c] * nrm);
}

// h = relu(out1 + b1), in place
__global__ __launch_bounds__(256) void k_bias_relu1(float* __restrict__ h,
                                                    const float* __restrict__ b1, int n) {
    int idx = blockIdx.x * blockDim.x + threadIdx.x;
    if ((idx >> 6) < n) {
        float v = h[idx] + b1[idx & 63];
        h[idx] = v > 0.0f ? v : 0.0f;
    }
}

// ---------------- GEMM 2: h2[M,48] = h[M,64] @ W2pad[64,48] ----------------
// W2 is 64x40 row-major; cols 40..47 are zero-padded via select (no EXEC
// divergence around the WMMA).

__global__ __launch_bounds__(256) void k_gemm2_wmma(const float* __restrict__ h,
                                                    const float* __restrict__ W2,
                                                    float* __restrict__ h2, int M) {
    const int NT = OUT_P / 16;                         // 3 col tiles
    int wave = blockIdx.x * (blockDim.x >> 5) + (threadIdx.x >> 5);
    int lane = threadIdx.x & 31;
    int mt = wave / NT, nt = wave % NT;
    if (mt * 16 >= M) return;

    int row  = mt * 16 + (lane & 15);
    int kg   = (lane >> 4) * 2;
    int ncol = nt * 16 + (lane & 15);
    bool valid = ncol < OUT_C;

    v8f acc = {};
    #pragma unroll 4
    for (int k0 = 0; k0 < HID_C; k0 += 4) {
        v2f a = *(const v2f*)(h + (size_t)row * HID_C + k0 + kg);
        v2f b;
        b.x = valid ? W2[(size_t)(k0 + kg + 0) * OUT_C + ncol] : 0.0f;
        b.y = valid ? W2[(size_t)(k0 + kg + 1) * OUT_C + ncol] : 0.0f;
        acc = __builtin_amdgcn_wmma_f32_16x16x4_f32(
                  false, a, false, b, (short)0, acc, false, false);
    }
    int half = (lane >> 4) * 8;
    #pragma unroll
    for (int v = 0; v < 8; ++v)
        h2[(size_t)(mt * 16 + half + v) * OUT_P + nt * 16 + (lane & 15)] = acc[v];
}

// ---------------- layer-2 aggregation ----------------

__global__ __launch_bounds__(256) void k_selfloop2(const float* __restrict__ h2,
                                                   const float* __restrict__ dinv,
                                                   float* __restrict__ out, int n) {
    int idx = blockIdx.x * blockDim.x + threadIdx.x;
    int i = idx / OUT_C, c = idx % OUT_C;
    if (i < n) {
        float d = dinv[i];
        out[idx] = h2[(size_t)i * OUT_P + c] * d * d;
    }
}

__global__ __launch_bounds__(256) void k_agg2(const float* __restrict__ h2,
                                              const int* __restrict__ src,
                                              const int* __restrict__ dst,
                                              const float* __restrict__ dinv,
                                              float* __restrict__ out, int e) {
    unsigned idx = blockIdx.x * blockDim.x + threadIdx.x;
    int ed = idx / OUT_C, c = idx % OUT_C;
    if (ed >= e) return;
    int s = src[ed], d = dst[ed];
    float nrm = dinv[s] * dinv[d];
    atomicAdd(out + (size_t)d * OUT_C + c, h2[(size_t)s * OUT_P + c] * nrm);
}

// ---------------- softmax (+b2), one wave32 per node, in place ----------------

__global__ __launch_bounds__(256) void k_softmax(float* __restrict__ out,
                                                 const float* __restrict__ b2, int n) {
    int wave = blockIdx.x * (blockDim.x >> 5) + (threadIdx.x >> 5);
    int lane = threadIdx.x & 31;
    if (wave >= n) return;
    float* p = out + (size_t)wave * OUT_C;

    float v0 = p[lane] + b2[lane];                           // lanes 0..31
    float v1 = (lane < OUT_C - 32) ? p[32 + lane] + b2[32 + lane] : -INFINITY;
    float m = fmaxf(v0, v1);
    #pragma unroll
    for (int o = 16; o > 0; o >>= 1) m = fmaxf(m, __shfl_xor(m, o, 32));

    float e0 = __expf(v0 - m);
    float e1 = (lane < OUT_C - 32) ? __expf(v1 - m) : 0.0f;
    float s = e0 + e1;
    #pragma unroll
    for (int o = 16; o > 0; o >>= 1) s += __shfl_xor(s, o, 32);

    float inv = 1.0f / s;
    p[lane] = e0 * inv;
    if (lane < OUT_C - 32) p[32 + lane] = e1 * inv;
}

// ---------------------------------------------------------------------------

extern "C" void kernel_launch(void* const* d_in, const int* in_sizes, int n_in,
                              void* d_out, int out_size, void* d_ws, size_t ws_size,
                              hipStream_t stream) {
    const float* x   = (const float*)d_in[0];
    const int*   ei  = (const int*)d_in[1];     // [2,E] flat: first E = src, next E = dst
    const float* W1  = (const float*)d_in[2];
    const float* b1  = (const float*)d_in[3];
    const float* W2  = (const float*)d_in[4];
    const float* b2  = (const float*)d_in[5];
    float* out = (float*)d_out;

    const int N = in_sizes[0] / IN_C;           // 50000
    const int E = in_sizes[1] / 2;              // 800000
    const int* src = ei;
    const int* dst = ei + E;

    // workspace layout (floats): dinv[N] | hlin[N*64] | hbuf[N*64]
    // hlin region is reused for h2 (N*48 <= N*64) after h is finalized in hbuf.
    float* dinv = (float*)d_ws;
    float* hlin = dinv + ((N + 255) & ~255);
    float* hbuf = hlin + (size_t)N * HID_C;
    float* h2   = hlin;                         // reuse

    const int B = 256;
    // 1) symmetric normalization
    k_init_deg   <<<(N + B - 1) / B, B, 0, stream>>>(dinv, N);
    k_deg_edges  <<<(E + B - 1) / B, B, 0, stream>>>(dst, dinv, E);
    k_deg_to_dinv<<<(N + B - 1) / B, B, 0, stream>>>(dinv, N);

    // 2) hlin = x @ W1   (WMMA f32)
    {
        int waves = ((N + 15) / 16) * (HID_C / 16);      // 3125*4
        int blocks = (waves + 7) / 8;                    // 8 waves/block
        k_gemm1_wmma<<<blocks, B, 0, stream>>>(x, W1, hlin, N);
    }

    // 3) aggregate layer 1, bias+relu -> h (in hbuf)
    k_selfloop1 <<<((size_t)N * HID_C + B - 1) / B, B, 0, stream>>>(hlin, dinv, hbuf, N);
    k_agg1      <<<((size_t)E * HID_C + B - 1) / B, B, 0, stream>>>(hlin, src, dst, dinv, hbuf, E);
    k_bias_relu1<<<((size_t)N * HID_C + B - 1) / B, B, 0, stream>>>(hbuf, b1, N);

    // 4) h2 = h @ W2   (WMMA f32, padded 40->48)
    {
        int waves = ((N + 15) / 16) * (OUT_P / 16);      // 3125*3
        int blocks = (waves + 7) / 8;
        k_gemm2_wmma<<<blocks, B, 0, stream>>>(hbuf, W2, h2, N);
    }

    // 5) aggregate layer 2 into d_out, then softmax(+b2) in place
    k_selfloop2<<<((size_t)N * OUT_C + B - 1) / B, B, 0, stream>>>(h2, dinv, out, N);
    k_agg2     <<<((size_t)E * OUT_C + B - 1) / B, B, 0, stream>>>(h2, src, dst, dinv, out, E);
    k_softmax  <<<(N * 32 + B - 1) / B, B, 0, stream>>>(out, b2, N);
}